// NodeEncoder_72997264163105
// MI455X (gfx1250) — compile-verified
//
#include <hip/hip_runtime.h>
#include <math.h>

#define N_NODES 50000
#define N_EDGES 640000
#define IN_DIM  64
#define HID     128
#define LN_EPS  1e-5f

typedef float v2f __attribute__((ext_vector_type(2)));
typedef float v8f __attribute__((ext_vector_type(8)));

// ---------------------------------------------------------------------------
// Y[N,128] = X[N,K] @ W[K,128] + bias   via V_WMMA_F32_16X16X4_F32
// One wave per 16x16 output tile; block = 256 threads = 8 waves = full 128-wide
// output row-of-tiles; grid.x = N/16 (50000/16 = 3125 exactly).
// A layout (16x4 f32): lanes 0-15 -> K=kb+0/kb+1, lanes 16-31 -> K=kb+2/kb+3.
// B layout (4x16 f32): mirrored; C/D: vgpr r = row r (lanes 0-15) / r+8 (16-31).
// ---------------------------------------------------------------------------
__global__ void gemm_xw_wmma(const float* __restrict__ X,
                             const float* __restrict__ W,
                             const float* __restrict__ bias,
                             float* __restrict__ Y,
                             int K) {
  const int lane = threadIdx.x & 31;
  const int wv   = threadIdx.x >> 5;       // 0..7 : N tile
  const int m0   = blockIdx.x * 16;
  const int n0   = wv * 16;
  const int half = lane >> 4;              // 0 for lanes 0-15, 1 for 16-31
  const int l    = lane & 15;

  v8f acc = {};
  const float* xrow = X + (size_t)(m0 + l) * K;
  for (int kb = 0; kb < K; kb += 4) {
    const int ka = kb + 2 * half;
    v2f a, b;
    a.x = xrow[ka];
    a.y = xrow[ka + 1];
    b.x = W[(size_t)ka * HID + n0 + l];
    b.y = W[(size_t)(ka + 1) * HID + n0 + l];
    acc = __builtin_amdgcn_wmma_f32_16x16x4_f32(false, a, false, b,
                                                (short)0, acc, false, false);
  }
  const float bb = bias ? bias[n0 + l] : 0.0f;
#pragma unroll
  for (int r = 0; r < 8; ++r) {
    const int row = m0 + r + 8 * half;
    Y[(size_t)row * HID + n0 + l] = acc[r] + bb;
  }
}

// ---------------------------------------------------------------------------
// segment-softmax helpers: monotonic float<->uint encoding for atomicMax
// ---------------------------------------------------------------------------
__device__ __forceinline__ unsigned fenc(float f) {
  unsigned u = __float_as_uint(f);
  return u ^ (((int)u < 0) ? 0xFFFFFFFFu : 0x80000000u);
}
__device__ __forceinline__ float fdec(unsigned u) {
  return __uint_as_float(u ^ (((int)u < 0) ? 0x80000000u : 0xFFFFFFFFu));
}

__global__ void init_maxsum(unsigned* __restrict__ amax,
                            float* __restrict__ asum, int nh) {
  int t = blockIdx.x * blockDim.x + threadIdx.x;
  if (t < nh) { amax[t] = fenc(-INFINITY); asum[t] = 0.0f; }
}

__global__ void zero_f32(float* __restrict__ p, int n) {
  int t = blockIdx.x * blockDim.x + threadIdx.x;
  if (t < n) p[t] = 0.0f;
}

// pass A: alpha[e,h] = dot(q[dst], k[src] + ea*We) * rscale ; amax = segmax
__global__ void edge_alpha(const int* __restrict__ ei,
                           const float* __restrict__ ea,
                           const float* __restrict__ Q,
                           const float* __restrict__ Km,
                           const float* __restrict__ We,
                           float* __restrict__ alpha,
                           unsigned* __restrict__ amax,
                           int H, int C, float rscale) {
  int t = blockIdx.x * blockDim.x + threadIdx.x;
  if (t >= N_EDGES * H) return;
  const int e = t / H, h = t - e * H;
  const int s = ei[e], d = ei[N_EDGES + e];
  const float w = ea[e];
  const float* q  = Q  + (size_t)d * HID + h * C;
  const float* k  = Km + (size_t)s * HID + h * C;
  const float* ev = We + h * C;
  float acc = 0.0f;
  for (int c = 0; c < C; ++c) acc = fmaf(q[c], k[c] + w * ev[c], acc);
  acc *= rscale;
  alpha[t] = acc;
  atomicMax(&amax[(size_t)d * H + h], fenc(acc));
}

// pass B: alpha <- exp(alpha - amax[dst]); asum = segsum
__global__ void edge_exp(const int* __restrict__ ei,
                         float* __restrict__ alpha,
                         const unsigned* __restrict__ amax,
                         float* __restrict__ asum, int H) {
  int t = blockIdx.x * blockDim.x + threadIdx.x;
  if (t >= N_EDGES * H) return;
  const int e = t / H, h = t - e * H;
  const int d = ei[N_EDGES + e];
  const float mx = fdec(amax[(size_t)d * H + h]);
  const float w  = expf(alpha[t] - mx);
  alpha[t] = w;
  atomicAdd(&asum[(size_t)d * H + h], w);
}

// pass C: out[dst] += (v[src] + ea*We) * alpha / (asum[dst] + 1e-16)
__global__ void edge_scatter(const int* __restrict__ ei,
                             const float* __restrict__ ea,
                             const float* __restrict__ V,
                             const float* __restrict__ We,
                             const float* __restrict__ alpha,
                             const float* __restrict__ asum,
                             float* __restrict__ out, int H, int C) {
  int t = blockIdx.x * blockDim.x + threadIdx.x;
  if (t >= N_EDGES * H) return;
  const int e = t / H, h = t - e * H;
  const int s = ei[e], d = ei[N_EDGES + e];
  const float w = ea[e];
  const float coeff = alpha[t] / (asum[(size_t)d * H + h] + 1e-16f);
  const float* v  = V  + (size_t)s * HID + h * C;
  const float* ev = We + h * C;
  float* o = out + (size_t)d * HID + h * C;
  for (int c = 0; c < C; ++c)
    atomicAdd(&o[c], (v[c] + w * ev[c]) * coeff);
}

// beta gate: one wave32 per node
__global__ void beta_gate(const float* __restrict__ out,
                          const float* __restrict__ Xr,
                          const float* __restrict__ Wb,
                          float* __restrict__ Xn, int relu_flag) {
  const int node = blockIdx.x * (blockDim.x >> 5) + (threadIdx.x >> 5);
  const int lane = threadIdx.x & 31;
  if (node >= N_NODES) return;
  const float* o = out + (size_t)node * HID;
  const float* r = Xr  + (size_t)node * HID;
  float bl = 0.0f;
  for (int j = lane; j < HID; j += 32) {
    const float ov = o[j], rv = r[j];
    bl += Wb[j] * ov + Wb[HID + j] * rv + Wb[2 * HID + j] * (ov - rv);
  }
  for (int off = 16; off > 0; off >>= 1) bl += __shfl_xor(bl, off, 32);
  const float beta = 1.0f / (1.0f + expf(-bl));
  float* xn = Xn + (size_t)node * HID;
  for (int j = lane; j < HID; j += 32) {
    float v = beta * r[j] + (1.0f - beta) * o[j];
    if (relu_flag) v = fmaxf(v, 0.0f);
    xn[j] = v;
  }
}

// LayerNorm + global mean pool: one wave32 per node, atomic accumulate / N
__global__ void ln_pool(const float* __restrict__ X,
                        const float* __restrict__ gamma,
                        const float* __restrict__ betap,
                        float* __restrict__ outp) {
  const int node = blockIdx.x * (blockDim.x >> 5) + (threadIdx.x >> 5);
  const int lane = threadIdx.x & 31;
  if (node >= N_NODES) return;
  const float* x = X + (size_t)node * HID;
  float s = 0.0f;
  for (int j = lane; j < HID; j += 32) s += x[j];
  for (int off = 16; off > 0; off >>= 1) s += __shfl_xor(s, off, 32);
  const float mu = s * (1.0f / HID);
  float var = 0.0f;
  for (int j = lane; j < HID; j += 32) { const float d = x[j] - mu; var += d * d; }
  for (int off = 16; off > 0; off >>= 1) var += __shfl_xor(var, off, 32);
  const float rstd  = rsqrtf(var * (1.0f / HID) + LN_EPS);
  const float inv_n = 1.0f / (float)N_NODES;
  for (int j = lane; j < HID; j += 32) {
    const float y = (x[j] - mu) * rstd * gamma[j] + betap[j];
    atomicAdd(&outp[j], y * inv_n);
  }
}

// ---------------------------------------------------------------------------
extern "C" void kernel_launch(void* const* d_in, const int* in_sizes, int n_in,
                              void* d_out, int out_size, void* d_ws, size_t ws_size,
                              hipStream_t stream) {
  (void)n_in; (void)out_size; (void)ws_size;
  const float* node_features = (const float*)d_in[0];
  const int*   edge_index    = (const int*)d_in[1];
  const float* edge_attr     = (const float*)d_in[2];

  const float *Wq[3], *bq[3], *Wk[3], *bk[3], *Wv[3], *bv[3];
  const float *We[3], *Ws[3], *bs[3], *Wb[3];
  const float *ln_gamma, *ln_beta;

  if (in_sizes[3] == 384) {
    // jax tree_flatten (sorted keys): Wb,We,Wk,Wq,Ws,Wv,bk,bq,bs,bv per layer,
    // then ln_beta, ln_gamma
    int i = 3;
    for (int l = 0; l < 3; ++l) {
      Wb[l] = (const float*)d_in[i + 0]; We[l] = (const float*)d_in[i + 1];
      Wk[l] = (const float*)d_in[i + 2]; Wq[l] = (const float*)d_in[i + 3];
      Ws[l] = (const float*)d_in[i + 4]; Wv[l] = (const float*)d_in[i + 5];
      bk[l] = (const float*)d_in[i + 6]; bq[l] = (const float*)d_in[i + 7];
      bs[l] = (const float*)d_in[i + 8]; bv[l] = (const float*)d_in[i + 9];
      i += 10;
    }
    ln_beta  = (const float*)d_in[i + 0];
    ln_gamma = (const float*)d_in[i + 1];
  } else {
    // insertion order: Wq,bq,Wk,bk,Wv,bv,We,Ws,bs,Wb per layer, then ln_gamma, ln_beta
    int i = 3;
    for (int l = 0; l < 3; ++l) {
      Wq[l] = (const float*)d_in[i + 0]; bq[l] = (const float*)d_in[i + 1];
      Wk[l] = (const float*)d_in[i + 2]; bk[l] = (const float*)d_in[i + 3];
      Wv[l] = (const float*)d_in[i + 4]; bv[l] = (const float*)d_in[i + 5];
      We[l] = (const float*)d_in[i + 6]; Ws[l] = (const float*)d_in[i + 7];
      bs[l] = (const float*)d_in[i + 8]; Wb[l] = (const float*)d_in[i + 9];
      i += 10;
    }
    ln_gamma = (const float*)d_in[i + 0];
    ln_beta  = (const float*)d_in[i + 1];
  }

  // workspace carve (~152 MB total)
  char* w = (char*)d_ws;
  auto take = [&](size_t bytes) {
    void* p = (void*)w;
    w += (bytes + 255) & ~(size_t)255;
    return p;
  };
  const size_t nodeMat = (size_t)N_NODES * HID * sizeof(float);
  float*    bufX  = (float*)take(nodeMat);
  float*    bufQ  = (float*)take(nodeMat);   // recycled as message accumulator
  float*    bufK  = (float*)take(nodeMat);
  float*    bufV  = (float*)take(nodeMat);
  float*    bufXr = (float*)take(nodeMat);
  float*    alpha = (float*)take((size_t)N_EDGES * 8 * sizeof(float));
  unsigned* amax  = (unsigned*)take((size_t)N_NODES * 8 * sizeof(unsigned));
  float*    asum  = (float*)take((size_t)N_NODES * 8 * sizeof(float));

  const int HEADS[3] = {8, 8, 1};
  const float* x = node_features;

  for (int l = 0; l < 3; ++l) {
    const int K = (l == 0) ? IN_DIM : HID;
    const int H = HEADS[l];
    const int C = HID / H;
    const float rscale = 1.0f / sqrtf((float)C);

    // dense projections: Q, K, V, Xr  (WMMA f32)
    dim3 gg(N_NODES / 16), gb(256);
    hipLaunchKernelGGL(gemm_xw_wmma, gg, gb, 0, stream, x, Wq[l], bq[l], bufQ,  K);
    hipLaunchKernelGGL(gemm_xw_wmma, gg, gb, 0, stream, x, Wk[l], bk[l], bufK,  K);
    hipLaunchKernelGGL(gemm_xw_wmma, gg, gb, 0, stream, x, Wv[l], bv[l], bufV,  K);
    hipLaunchKernelGGL(gemm_xw_wmma, gg, gb, 0, stream, x, Ws[l], bs[l], bufXr, K);

    const int nh  = N_NODES * H;
    const int eh  = N_EDGES * H;
    dim3 b256(256);
    hipLaunchKernelGGL(init_maxsum, dim3((nh + 255) / 256), b256, 0, stream,
                       amax, asum, nh);
    hipLaunchKernelGGL(edge_alpha, dim3((eh + 255) / 256), b256, 0, stream,
                       edge_index, edge_attr, bufQ, bufK, We[l],
                       alpha, amax, H, C, rscale);
    // Q no longer needed -> recycle bufQ as the scatter accumulator
    hipLaunchKernelGGL(zero_f32, dim3((N_NODES * HID + 255) / 256), b256, 0, stream,
                       bufQ, N_NODES * HID);
    hipLaunchKernelGGL(edge_exp, dim3((eh + 255) / 256), b256, 0, stream,
                       edge_index, alpha, amax, asum, H);
    hipLaunchKernelGGL(edge_scatter, dim3((eh + 255) / 256), b256, 0, stream,
                       edge_index, edge_attr, bufV, We[l], alpha, asum,
                       bufQ, H, C);
    // beta gate (+ReLU for layers 0,1) -> bufX
    hipLaunchKernelGGL(beta_gate, dim3((N_NODES + 7) / 8), b256, 0, stream,
                       bufQ, bufXr, Wb[l], bufX, (l < 2) ? 1 : 0);
    x = bufX;
  }

  // LayerNorm + global mean pool into d_out[1,128]
  hipLaunchKernelGGL(zero_f32, dim3(1), dim3(128), 0, stream, (float*)d_out, HID);
  hipLaunchKernelGGL(ln_pool, dim3((N_NODES + 7) / 8), dim3(256), 0, stream,
                     bufX, ln_gamma, ln_beta, (float*)d_out);
}